// StupidIndexator_68934225101098
// MI455X (gfx1250) — compile-verified
//
#include <hip/hip_runtime.h>

typedef __attribute__((ext_vector_type(2))) float v2f;
typedef __attribute__((ext_vector_type(8))) float v8f;

#define N_OMEGA 16384
#define N_UF    128
#define N_HKL   256
// cos(0.02) to full f32 precision
#define COS_THR 0.99980000666657784f

// One wave32 per omega; 8 waves (256 threads) per block.
// Per omega: 16 N-tiles x 8 M-tiles of V_WMMA_F32_16X16X4_F32 (K=3 padded to 4).
__global__ __launch_bounds__(256) void rate_kernel(
    const float* __restrict__ uf,
    const float* __restrict__ omega,
    const float* __restrict__ recip,
    const float* __restrict__ hkl,
    float* __restrict__ rate)
{
    __shared__ float s_uq[N_UF * 3];    // normalized (uf - u_i)
    __shared__ float s_hkl[N_HKL * 3];

    const int tid = threadIdx.x;

    // Stage uq_exp = normalize(uf - (0,0,1)) into LDS (threads 0..127)
    if (tid < N_UF) {
        float x = uf[tid * 3 + 0];
        float y = uf[tid * 3 + 1];
        float z = uf[tid * 3 + 2] - 1.0f;
        float inv = rsqrtf(x * x + y * y + z * z);
        s_uq[tid * 3 + 0] = x * inv;
        s_uq[tid * 3 + 1] = y * inv;
        s_uq[tid * 3 + 2] = z * inv;
    }
    // Stage hkl (threads 0..255, one row each)
    s_hkl[tid * 3 + 0] = hkl[tid * 3 + 0];
    s_hkl[tid * 3 + 1] = hkl[tid * 3 + 1];
    s_hkl[tid * 3 + 2] = hkl[tid * 3 + 2];
    __syncthreads();

    const int wave = tid >> 5;
    const int lane = tid & 31;
    const int l16  = lane & 15;
    const bool lo  = (lane < 16);
    const int c    = blockIdx.x * 8 + wave;

    // ---- rotation R = Rx(w1) Ry(w2) Rz(w3) ----
    float w1 = omega[c * 3 + 0];
    float w2 = omega[c * 3 + 1];
    float w3 = omega[c * 3 + 2];
    float s1, c1, s2, c2, s3, c3;
    sincosf(w1, &s1, &c1);
    sincosf(w2, &s2, &c2);
    sincosf(w3, &s3, &c3);
    // R rows (Rx * Ry * Rz expanded analytically)
    float R00 =  c2 * c3,                R01 = -c2 * s3,                R02 =  s2;
    float R10 =  c1 * s3 + s1 * s2 * c3, R11 =  c1 * c3 - s1 * s2 * s3, R12 = -s1 * c2;
    float R20 =  s1 * s3 - c1 * s2 * c3, R21 =  s1 * c3 + c1 * s2 * s3, R22 =  c1 * c2;

    // rec = R @ reciprocal
    float G00 = recip[0], G01 = recip[1], G02 = recip[2];
    float G10 = recip[3], G11 = recip[4], G12 = recip[5];
    float G20 = recip[6], G21 = recip[7], G22 = recip[8];
    float r00 = R00 * G00 + R01 * G10 + R02 * G20;
    float r01 = R00 * G01 + R01 * G11 + R02 * G21;
    float r02 = R00 * G02 + R01 * G12 + R02 * G22;
    float r10 = R10 * G00 + R11 * G10 + R12 * G20;
    float r11 = R10 * G01 + R11 * G11 + R12 * G21;
    float r12 = R10 * G02 + R11 * G12 + R12 * G22;
    float r20 = R20 * G00 + R21 * G10 + R22 * G20;
    float r21 = R20 * G01 + R21 * G11 + R22 * G21;
    float r22 = R20 * G02 + R21 * G12 + R22 * G22;

    // ---- Preload 8 A-tiles (uq_exp): A is 16x4 f32, 2 VGPRs/lane.
    // VGPR0: lanes 0-15 K=0 (x), lanes 16-31 K=2 (z)
    // VGPR1: lanes 0-15 K=1 (y), lanes 16-31 K=3 (pad 0)
    v2f A[8];
#pragma unroll
    for (int t = 0; t < 8; ++t) {
        int e = t * 16 + l16;
        A[t].x = lo ? s_uq[e * 3 + 0] : s_uq[e * 3 + 2];
        A[t].y = lo ? s_uq[e * 3 + 1] : 0.0f;
    }

    // per-lane running max|dot| for this lane's m-columns; [t*8+r] -> e = 16t + r (+8 hi half)
    float vmax[64];
#pragma unroll
    for (int k = 0; k < 64; ++k) vmax[k] = 0.0f;

    // ---- main loop over 16 N-tiles of hkl ----
    for (int n = 0; n < 16; ++n) {
        int m = n * 16 + l16;
        float h0 = s_hkl[m * 3 + 0];
        float h1 = s_hkl[m * 3 + 1];
        float h2 = s_hkl[m * 3 + 2];
        float qx = r00 * h0 + r01 * h1 + r02 * h2;
        float qy = r10 * h0 + r11 * h1 + r12 * h2;
        float qz = r20 * h0 + r21 * h1 + r22 * h2;
        float inv = rsqrtf(qx * qx + qy * qy + qz * qz);
        // B 4x16: VGPR0: lanes 0-15 K=0 (x), lanes 16-31 K=2 (z); VGPR1: K=1 (y) / K=3 (0)
        v2f B;
        B.x = (lo ? qx : qz) * inv;
        B.y = lo ? (qy * inv) : 0.0f;

#pragma unroll
        for (int t = 0; t < 8; ++t) {
            v8f acc = {0.0f, 0.0f, 0.0f, 0.0f, 0.0f, 0.0f, 0.0f, 0.0f};
            acc = __builtin_amdgcn_wmma_f32_16x16x4_f32(
                /*neg_a=*/false, A[t], /*neg_b=*/false, B,
                /*c_mod=*/(short)0, acc, /*reuse_a=*/false, /*reuse_b=*/false);
            // max(v, |d|) == max3(v, d, -d): single V_MAX3_NUM_F32 with neg modifier
#pragma unroll
            for (int r = 0; r < 8; ++r)
                vmax[t * 8 + r] = fmaxf(fmaxf(vmax[t * 8 + r], acc[r]), -acc[r]);
        }
    }

    // ---- threshold first, then OR-reduce bitmasks across the 16-lane half-groups.
    // max over lanes >= thr  <=>  OR over lanes of (per-lane max >= thr), so this is
    // exact and replaces 64x4 float shuffles with 2x4 mask shuffles.
    unsigned int b0 = 0u, b1 = 0u;
#pragma unroll
    for (int k = 0; k < 32; ++k)
        b0 |= (vmax[k] >= COS_THR) ? (1u << k) : 0u;
#pragma unroll
    for (int k = 0; k < 32; ++k)
        b1 |= (vmax[32 + k] >= COS_THR) ? (1u << k) : 0u;

#pragma unroll
    for (int mask = 1; mask < 16; mask <<= 1) {
        b0 |= __shfl_xor(b0, mask, 32);
        b1 |= __shfl_xor(b1, mask, 32);
    }
    // bit (t*8+r) of {b0,b1} == matched for e = 16t + r (low lanes) / 16t + 8 + r (high lanes)
    int cnt = __popc(b0) + __popc(b1);
    cnt += __shfl_xor(cnt, 16, 32);   // merge e%16 in 0..7 with e%16 in 8..15

    if (lane == 0) rate[c] = (float)cnt;
}

// Deterministic, stable descending rank (matches stable jnp.argsort(-rate)):
// pos[i] = #{j: rate_j > rate_i} + #{j < i: rate_j == rate_i}. Rates are exact
// small integers in f32, so == is exact. O(n^2) with LDS tiling: 268M compares.
__global__ __launch_bounds__(256) void rank_scatter_kernel(
    const float* __restrict__ omega,
    const float* __restrict__ rate,
    float* __restrict__ out)
{
    __shared__ float tile[256];
    const int i = blockIdx.x * 256 + threadIdx.x;
    const float ri = rate[i];
    int gt = 0, eqb = 0;
    for (int base = 0; base < N_OMEGA; base += 256) {
        __syncthreads();
        tile[threadIdx.x] = rate[base + threadIdx.x];
        __syncthreads();
#pragma unroll 8
        for (int jj = 0; jj < 256; ++jj) {
            float rj = tile[jj];
            int j = base + jj;
            gt  += (rj > ri) ? 1 : 0;
            eqb += ((rj == ri) && (j < i)) ? 1 : 0;
        }
    }
    const int pos = gt + eqb;
    out[pos * 3 + 0] = omega[i * 3 + 0];
    out[pos * 3 + 1] = omega[i * 3 + 1];
    out[pos * 3 + 2] = omega[i * 3 + 2];
    out[N_OMEGA * 3 + pos] = ri;
}

extern "C" void kernel_launch(void* const* d_in, const int* in_sizes, int n_in,
                              void* d_out, int out_size, void* d_ws, size_t ws_size,
                              hipStream_t stream) {
    const float* uf    = (const float*)d_in[0];  // (128,3)
    const float* omega = (const float*)d_in[1];  // (16384,3)
    const float* recip = (const float*)d_in[2];  // (3,3)
    const float* hkl   = (const float*)d_in[3];  // (256,3)
    float* rate = (float*)d_ws;                  // 16384 floats scratch
    float* out  = (float*)d_out;                 // 16384*3 omega + 16384 rate

    rate_kernel<<<N_OMEGA / 8, 256, 0, stream>>>(uf, omega, recip, hkl, rate);
    rank_scatter_kernel<<<N_OMEGA / 256, 256, 0, stream>>>(omega, rate, out);
}